// LoRAElementCWLinear_17600775979363
// MI455X (gfx1250) — compile-verified
//
#include <hip/hip_runtime.h>

typedef float v2f __attribute__((ext_vector_type(2)));
typedef float v4f __attribute__((ext_vector_type(4)));
typedef float v8f __attribute__((ext_vector_type(8)));

#define NUM_ELEMENTS 10
#define NUM_PATH     4
#define CHANNEL      128
#define M_DIM        3
#define LORA_R       8
#define SCALING      1.0f
#define ALPHA_C      0.5f          /* 1/sqrt(NUM_PATH) */
#define ROW          (CHANNEL * M_DIM)                   /* 384 floats */
#define WTAB         (NUM_ELEMENTS * CHANNEL * NUM_PATH) /* 5120 floats, 20KB */

// ---------------------------------------------------------------------------
// Kernel 1: W_combined[z][c][a] = (weights[z][c][a] + delta[c][a]) * ALPHA
//   delta[c][a] = sum_r lora_B[r][c] * lora_A[a][r] * SCALING
// fp32 GEMM [128,8]x[8,4] via V_WMMA_F32_16X16X4_F32: 8 waves = 8 M-tiles of
// 16 channels, each wave does 2 accumulating K-chunks of 4 (LORA_R = 8).
// A 16x4 f32 (ISA 7.12.2): lanes 0-15 M=lane, VGPR0=K0, VGPR1=K1;
//                          lanes 16-31: VGPR0=K2, VGPR1=K3.
// B 4x16: same K split per lane half, N = lane%16 (zero-padded N>=4).
// D 16x16: VGPR j -> (M = j + 8*(lane/16), N = lane%16).
// ---------------------------------------------------------------------------
__global__ __launch_bounds__(256) void prep_weights_kernel(
    const float* __restrict__ weights, const float* __restrict__ lora_A,
    const float* __restrict__ lora_B, float* __restrict__ Wc)
{
  const int tid  = threadIdx.x;
  const int wave = tid >> 5;          // channel tile 0..7
  const int lane = tid & 31;
  const int M    = lane & 15;
  const int half = lane >> 4;
  const int ch   = wave * 16 + M;     // channel (A-matrix row)
  const int kloc = half * 2;          // local K pair held by this lane half
  const int n    = M;                 // N (path) column for B operand
  const int ns   = (n < NUM_PATH) ? n : 0;        // clamped: no OOB loads
  const float nvalid = (n < NUM_PATH) ? 1.0f : 0.0f;

  v8f acc = {};                       // C = 0
  #pragma unroll
  for (int ko = 0; ko < LORA_R; ko += 4) {
    v2f a, b;
    a.x = lora_B[(ko + kloc    ) * CHANNEL + ch];      // A[M][K]
    a.y = lora_B[(ko + kloc + 1) * CHANNEL + ch];
    b.x = lora_A[ns * LORA_R + ko + kloc    ] * SCALING * nvalid;  // B[K][N]
    b.y = lora_A[ns * LORA_R + ko + kloc + 1] * SCALING * nvalid;
    acc = __builtin_amdgcn_wmma_f32_16x16x4_f32(
        /*neg_a=*/false, a, /*neg_b=*/false, b,
        /*c_mod=*/(short)0, acc, /*reuse_a=*/false, /*reuse_b=*/false);
  }

  if (n < NUM_PATH) {
    #pragma unroll
    for (int j = 0; j < 8; ++j) {
      const int cr = wave * 16 + j + 8 * half;    // channel of D element j
      const float d = acc[j];                     // delta[cr][n]
      #pragma unroll
      for (int z = 0; z < NUM_ELEMENTS; ++z) {
        const int idx = (z * CHANNEL + cr) * NUM_PATH + n;
        Wc[idx] = (weights[idx] + d) * ALPHA_C;
      }
    }
  }
}

// ---------------------------------------------------------------------------
// Kernel 2: HBM-bound streaming contraction (~768 MB -> ~33 us floor).
// One wave per node; lane L owns channels 4L..4L+3: 12 contiguous floats per
// path -> 3 aligned b128 non-temporal loads per path + 3 NT stores.
// z from one-hot attrs row via wave32 ballot; 20 KB weight table in LDS,
// each lane's 16 weights are one contiguous 64 B LDS run (4x ds_load_b128).
// ---------------------------------------------------------------------------
__global__ __launch_bounds__(256) void lora_cw_linear_kernel(
    const float* __restrict__ t, const float* __restrict__ attrs,
    const float* __restrict__ Wc, float* __restrict__ out, int nodes)
{
  __shared__ float Wlds[WTAB];
  const int tid = threadIdx.x;
  for (int i = tid; i < WTAB; i += 256) Wlds[i] = Wc[i];
  __syncthreads();

  const int wave = tid >> 5;
  const int lane = tid & 31;
  const int stride = gridDim.x * 8;               // total waves in grid

  for (int node = blockIdx.x * 8 + wave; node < nodes; node += stride) {
    float av = (lane < NUM_ELEMENTS)
                   ? attrs[(size_t)node * NUM_ELEMENTS + lane] : 0.0f;
    unsigned mask = __builtin_amdgcn_ballot_w32(av > 0.5f);
    int z = mask ? (int)__builtin_ctz(mask) : 0;

    const v4f* wp = (const v4f*)(Wlds + z * (CHANNEL * NUM_PATH) + lane * 16);
    v4f w0 = wp[0], w1 = wp[1], w2 = wp[2], w3 = wp[3];
    float wv[16];
    #pragma unroll
    for (int j = 0; j < 4; ++j) {
      wv[j] = w0[j]; wv[4 + j] = w1[j]; wv[8 + j] = w2[j]; wv[12 + j] = w3[j];
    }

    float acc[12];
    #pragma unroll
    for (int j = 0; j < 12; ++j) acc[j] = 0.0f;

    #pragma unroll
    for (int a = 0; a < NUM_PATH; ++a) {
      const float* base = t + ((size_t)a * nodes + node) * ROW + lane * 12;
      v4f q0 = __builtin_nontemporal_load((const v4f*)(base));
      v4f q1 = __builtin_nontemporal_load((const v4f*)(base + 4));
      v4f q2 = __builtin_nontemporal_load((const v4f*)(base + 8));
      float v[12];
      #pragma unroll
      for (int j = 0; j < 4; ++j) { v[j] = q0[j]; v[4 + j] = q1[j]; v[8 + j] = q2[j]; }
      #pragma unroll
      for (int j = 0; j < 12; ++j)            // channel of float j is j/3
        acc[j] = fmaf(wv[(j / 3) * 4 + a], v[j], acc[j]);
    }

    int nnode = node + stride;                 // prefetch next node's rows
    if (nnode < nodes) {
      #pragma unroll
      for (int a = 0; a < NUM_PATH; ++a)
        __builtin_prefetch(t + ((size_t)a * nodes + nnode) * ROW + lane * 12, 0, 0);
    }

    float* ob = out + (size_t)node * ROW + lane * 12;
    v4f o0, o1, o2;
    #pragma unroll
    for (int j = 0; j < 4; ++j) { o0[j] = acc[j]; o1[j] = acc[4 + j]; o2[j] = acc[8 + j]; }
    __builtin_nontemporal_store(o0, (v4f*)(ob));
    __builtin_nontemporal_store(o1, (v4f*)(ob + 4));
    __builtin_nontemporal_store(o2, (v4f*)(ob + 8));
  }
}

extern "C" void kernel_launch(void* const* d_in, const int* in_sizes, int n_in,
                              void* d_out, int out_size, void* d_ws, size_t ws_size,
                              hipStream_t stream) {
  const float* t       = (const float*)d_in[0];
  const float* attrs   = (const float*)d_in[1];
  const float* weights = (const float*)d_in[2];
  const float* lora_A  = (const float*)d_in[3];
  const float* lora_B  = (const float*)d_in[4];
  float* out = (float*)d_out;
  float* Wc  = (float*)d_ws;                  // 20 KB scratch for combined W

  const int nodes = in_sizes[1] / NUM_ELEMENTS;

  prep_weights_kernel<<<1, 256, 0, stream>>>(weights, lora_A, lora_B, Wc);

  int nblk = (nodes + 7) / 8;
  if (nblk > 2048) nblk = 2048;               // persistent blocks, grid-stride
  lora_cw_linear_kernel<<<nblk, 256, 0, stream>>>(t, attrs, Wc, out, nodes);
}